// FactorizedRNN_2113123909736
// MI455X (gfx1250) — compile-verified
//
#include <hip/hip_runtime.h>
#include <math.h>

typedef __attribute__((ext_vector_type(2))) float v2f;
typedef __attribute__((ext_vector_type(8))) float v8f;

#define LDA_S 20   // 128 x 16 A tile, padded: stride-20 rows are bank-conflict free
#define LDB_S 136  // 16 x 128 B tile, padded so lane l and l+16 hit disjoint banks

// C[M,N] = op(A) @ op(B)  with f32 WMMA 16x16x4.
// TA: A is (K,M) accessed A[k][i]   else (M,K) A[i][k]
// TB: B is (N,K) accessed B[j][k]   else (K,N) B[k][j]
// EPI: 0 = C := G ; 1 = C += G ; 2 = fused FISTA shrink + momentum
template<bool TA, bool TB, int EPI>
__global__ __launch_bounds__(256) void gemm_wmma(
    const float* __restrict__ A, const float* __restrict__ B,
    float* __restrict__ C, int M, int N, int K,
    const float* __restrict__ XtW, const float* __restrict__ Yin,
    const float* __restrict__ Hcur, float* __restrict__ Hout,
    float* __restrict__ Yout, const float* __restrict__ Lptr,
    float mu, float lam) {
  __shared__ float sA[128 * LDA_S];
  __shared__ float sB[16 * LDB_S];

  const int tid   = threadIdx.x;
  const int lane  = tid & 31;
  const int lrow  = lane & 15;
  const int lhalf = lane >> 4;          // 0: K0/K1 half, 1: K2/K3 half
  const int wave  = tid >> 5;           // 8 waves
  const int wrow  = wave >> 2;          // 0..1 -> 64-row band
  const int wcol  = wave & 3;           // 0..3 -> 32-col band
  const int blkM  = blockIdx.y, blkN = blockIdx.x;
  const int lda   = TA ? M : K;
  const int ldb   = TB ? K : N;

  v8f acc[4][2];
  const v8f zero = {0.f,0.f,0.f,0.f,0.f,0.f,0.f,0.f};
#pragma unroll
  for (int mi = 0; mi < 4; ++mi)
#pragma unroll
    for (int ni = 0; ni < 2; ++ni) acc[mi][ni] = zero;

  for (int k0 = 0; k0 < K; k0 += 16) {
    // ---- stage A tile (128 rows x 16 k) into LDS ----
#pragma unroll
    for (int it = 0; it < 2; ++it) {
      int idx = tid + it * 256;
      if (TA) {   // A[k][i]: load row-major chunk, transpose into sA[i][k]
        int r = idx >> 5, cq = idx & 31;
        const float4 av = *(const float4*)(A + (size_t)(k0 + r) * lda + blkM * 128 + cq * 4);
        sA[(cq * 4 + 0) * LDA_S + r] = av.x;
        sA[(cq * 4 + 1) * LDA_S + r] = av.y;
        sA[(cq * 4 + 2) * LDA_S + r] = av.z;
        sA[(cq * 4 + 3) * LDA_S + r] = av.w;
      } else {    // A[i][k]
        int row = idx >> 2, q = idx & 3;
        const float4 av = *(const float4*)(A + (size_t)(blkM * 128 + row) * lda + k0 + q * 4);
        *(float4*)&sA[row * LDA_S + q * 4] = av;
      }
    }
    // ---- stage B tile (16 k x 128 cols) into LDS ----
#pragma unroll
    for (int it = 0; it < 2; ++it) {
      int idx = tid + it * 256;
      if (TB) {   // B[j][k]: transpose into sB[k][j]
        int nrow = idx >> 2, q = idx & 3;
        const float4 bv = *(const float4*)(B + (size_t)(blkN * 128 + nrow) * ldb + k0 + q * 4);
        sB[(q * 4 + 0) * LDB_S + nrow] = bv.x;
        sB[(q * 4 + 1) * LDB_S + nrow] = bv.y;
        sB[(q * 4 + 2) * LDB_S + nrow] = bv.z;
        sB[(q * 4 + 3) * LDB_S + nrow] = bv.w;
      } else {    // B[k][j]
        int r = idx >> 5, cq = idx & 31;
        const float4 bv = *(const float4*)(B + (size_t)(k0 + r) * ldb + blkN * 128 + cq * 4);
        *(float4*)&sB[r * LDB_S + cq * 4] = bv;
      }
    }
    __syncthreads();

    // ---- 4 WMMA k-steps of 4 over the 16-wide chunk ----
#pragma unroll
    for (int kk = 0; kk < 16; kk += 4) {
      v2f afrag[4];
#pragma unroll
      for (int mi = 0; mi < 4; ++mi) {
        const float* p = &sA[(wrow * 64 + mi * 16 + lrow) * LDA_S + kk + 2 * lhalf];
        afrag[mi] = *(const v2f*)p;    // lanes 0-15: K0,K1 ; 16-31: K2,K3
      }
      v2f bfrag[2];
#pragma unroll
      for (int ni = 0; ni < 2; ++ni) {
        const float* p = &sB[(kk + 2 * lhalf) * LDB_S + wcol * 32 + ni * 16 + lrow];
        v2f b; b.x = p[0]; b.y = p[LDB_S];
        bfrag[ni] = b;
      }
#pragma unroll
      for (int mi = 0; mi < 4; ++mi)
#pragma unroll
        for (int ni = 0; ni < 2; ++ni)
          acc[mi][ni] = __builtin_amdgcn_wmma_f32_16x16x4_f32(
              false, afrag[mi], false, bfrag[ni], (short)0, acc[mi][ni], false, false);
    }
    __syncthreads();
  }

  // ---- epilogue ----
  float invL = 0.f;
  if (EPI == 2) invL = 1.0f / Lptr[0];
#pragma unroll
  for (int mi = 0; mi < 4; ++mi)
#pragma unroll
    for (int ni = 0; ni < 2; ++ni)
#pragma unroll
      for (int v = 0; v < 8; ++v) {
        int r = blkM * 128 + wrow * 64 + mi * 16 + v + 8 * lhalf;
        int c = blkN * 128 + wcol * 32 + ni * 16 + lrow;
        size_t idx = (size_t)r * N + c;
        float g = acc[mi][ni][v];
        if (EPI == 0) {
          C[idx] = g;
        } else if (EPI == 1) {
          C[idx] += g;
        } else {
          float yv = Yin[idx];
          float hn = fmaxf(yv - (g - XtW[idx] + lam) * invL, 0.0f);
          float yn = hn + mu * (hn - Hcur[idx]);
          Hout[idx] = hn;
          Yout[idx] = yn;
        }
      }
}

__global__ __launch_bounds__(256) void init_ones_k(float* __restrict__ v, int n) {
  int i = blockIdx.x * 256 + threadIdx.x;
  if (i < n) v[i] = 1.0f;
}

// w = WtW @ v ; one wave32 per row
__global__ __launch_bounds__(256) void matvec_k(const float* __restrict__ Mtx,
                                                const float* __restrict__ v,
                                                float* __restrict__ w, int n) {
  int wv = threadIdx.x >> 5, lane = threadIdx.x & 31;
  int row = blockIdx.x * 8 + wv;
  const float* mr = Mtx + (size_t)row * n;
  float s = 0.f;
  for (int j = lane; j < n; j += 32) s += mr[j] * v[j];
  for (int off = 16; off > 0; off >>= 1) s += __shfl_xor(s, off, 32);
  if (lane == 0) w[row] = s;
}

// lambda = ||w|| ; v = w / lambda ; Lout = lambda + 1e-8
__global__ __launch_bounds__(256) void norm_scale_k(const float* __restrict__ w,
                                                    float* __restrict__ v,
                                                    float* __restrict__ Lout, int n) {
  __shared__ float red[8];
  __shared__ float lam_s;
  float s = 0.f;
  for (int j = threadIdx.x; j < n; j += 256) { float x = w[j]; s += x * x; }
  for (int off = 16; off > 0; off >>= 1) s += __shfl_xor(s, off, 32);
  int wv = threadIdx.x >> 5, lane = threadIdx.x & 31;
  if (lane == 0) red[wv] = s;
  __syncthreads();
  if (wv == 0) {
    float tot = (lane < 8) ? red[lane] : 0.f;
    for (int off = 16; off > 0; off >>= 1) tot += __shfl_xor(tot, off, 32);
    if (lane == 0) { float l = sqrtf(tot); lam_s = l; Lout[0] = l + 1e-8f; }
  }
  __syncthreads();
  float rl = 1.0f / lam_s;
  for (int j = threadIdx.x; j < n; j += 256) v[j] = w[j] * rl;
}

extern "C" void kernel_launch(void* const* d_in, const int* in_sizes, int n_in,
                              void* d_out, int out_size, void* d_ws, size_t ws_size,
                              hipStream_t stream) {
  (void)in_sizes; (void)n_in; (void)out_size; (void)ws_size;
  const float* x      = (const float*)d_in[0];  // (2048,1024)
  const float* h_prev = (const float*)d_in[1];  // (2048,2048)
  const float* W_y    = (const float*)d_in[2];  // (1024,2048)
  const float* W_h    = (const float*)d_in[3];  // (2048,2048)
  const float* W_x    = (const float*)d_in[4];  // (1024,2048)
  float* out = (float*)d_out;                   // [y (2048x1024) | h (2048x2048)]

  const int Bm = 2048, HD = 2048, XD = 1024, YD = 1024;
  const size_t S = (size_t)HD * HD;

  float* ws   = (float*)d_ws;
  float* WtW  = ws;
  float* XtW  = ws + S;
  float* Hb[2] = { ws + 2 * S, ws + 3 * S };
  float* Yb[2] = { ws + 4 * S, ws + 5 * S };
  float* vvec = ws + 6 * S;
  float* wvec = vvec + HD;
  float* Lptr = wvec + HD;

  const dim3 blk(256);
  const dim3 g16(HD / 128, HD / 128);
  const float lam = 0.1f;

  // WtW = W_h^T W_h + W_x^T W_x   (K split over the concat)
  gemm_wmma<true,false,0><<<g16, blk, 0, stream>>>(W_h, W_h, WtW, HD, HD, HD,
      nullptr, nullptr, nullptr, nullptr, nullptr, nullptr, 0.f, 0.f);
  gemm_wmma<true,false,1><<<g16, blk, 0, stream>>>(W_x, W_x, WtW, HD, HD, XD,
      nullptr, nullptr, nullptr, nullptr, nullptr, nullptr, 0.f, 0.f);

  // XtW = h_prev @ W_h + x @ W_x
  gemm_wmma<false,false,0><<<g16, blk, 0, stream>>>(h_prev, W_h, XtW, Bm, HD, HD,
      nullptr, nullptr, nullptr, nullptr, nullptr, nullptr, 0.f, 0.f);
  gemm_wmma<false,false,1><<<g16, blk, 0, stream>>>(x, W_x, XtW, Bm, HD, XD,
      nullptr, nullptr, nullptr, nullptr, nullptr, nullptr, 0.f, 0.f);

  // Power iteration for L = lambda_max(WtW) + 1e-8 (positive matrix -> fast convergence)
  init_ones_k<<<HD / 256, blk, 0, stream>>>(vvec, HD);
  for (int i = 0; i < 20; ++i) {
    matvec_k<<<HD / 8, blk, 0, stream>>>(WtW, vvec, wvec, HD);
    norm_scale_k<<<1, blk, 0, stream>>>(wvec, vvec, Lptr, HD);
  }

  // H0 = Y0 = h_prev
  hipMemcpyAsync(Hb[0], h_prev, S * sizeof(float), hipMemcpyDeviceToDevice, stream);
  hipMemcpyAsync(Yb[0], h_prev, S * sizeof(float), hipMemcpyDeviceToDevice, stream);

  // 55 FISTA steps (momentum t resets at step 50, matching the two fista calls)
  double t = 1.0;
  for (int k = 0; k < 55; ++k) {
    if (k == 50) t = 1.0;
    double tn = 0.5 * (1.0 + sqrt(1.0 + 4.0 * t * t));
    float mu = (float)((t - 1.0) / tn);
    t = tn;
    int in = k & 1, o = in ^ 1;
    float* Hout = (k == 54) ? (out + (size_t)Bm * YD) : Hb[o];
    gemm_wmma<false,false,2><<<g16, blk, 0, stream>>>(Yb[in], WtW, nullptr, Bm, HD, HD,
        XtW, Yb[in], Hb[in], Hout, Yb[o], Lptr, mu, lam);
  }

  // y = h @ W_y^T
  const float* hfin = out + (size_t)Bm * YD;
  gemm_wmma<false,true,0><<<dim3(YD / 128, Bm / 128), blk, 0, stream>>>(hfin, W_y, out,
      Bm, YD, HD, nullptr, nullptr, nullptr, nullptr, nullptr, nullptr, 0.f, 0.f);
}